// snn_linear_41583873360168
// MI455X (gfx1250) — compile-verified
//
#include <hip/hip_runtime.h>
#include <hip/hip_bf16.h>

// SNN linear layer for MI455X (gfx1250, wave32).
// syn = spikes^T (1x4096) * W (4096x8192 f32): 128 MB single-use stream -> memory bound
// (~5.5 us at 23.3 TB/s). Split-K across 8 chunks (512 blocks total) for enough
// memory-level parallelism to saturate HBM; f32 WMMA (16x16x4) does the accumulation
// at full precision. Partials go to d_ws; a deterministic epilogue kernel reduces them
// (no FP atomics -> bitwise-reproducible across replays).

#define NEU_IN   4096
#define NEU_OUT  8192
#define K_CHUNKS 8
#define K_PER    (NEU_IN / K_CHUNKS)     // 512 rows per block
#define COLS_PER_BLOCK 128               // 8 waves x 16-column WMMA tiles

typedef float v2f __attribute__((ext_vector_type(2)));
typedef float v8f __attribute__((ext_vector_type(8)));
typedef int   v4i __attribute__((ext_vector_type(4)));

// Pointer types matching the async-LDS builtin's signature:
// (v4i addrspace(1)* src, v4i addrspace(3)* dst, imm offset, imm cpol)
typedef __attribute__((address_space(1))) v4i* gptr_v4i;
typedef __attribute__((address_space(3))) v4i* lptr_v4i;

#if defined(__has_builtin)
#if __has_builtin(__builtin_amdgcn_global_load_async_to_lds_b128)
#define HAVE_ASYNC_LDS 1
#else
#define HAVE_ASYNC_LDS 0
#endif
#else
#define HAVE_ASYNC_LDS 0
#endif

__global__ __launch_bounds__(256) void snn_partial_wmma_kernel(
    const int* __restrict__ spikes_in,   // [NEU_IN] 0/1
    const float* __restrict__ weights,   // [NEU_IN, NEU_OUT] row-major
    float* __restrict__ partial)         // [K_CHUNKS][NEU_OUT]
{
    __shared__ float s_spk[K_PER];       // 2 KB f32 spikes for this K chunk
    const int k0 = blockIdx.y * K_PER;

#if HAVE_ASYNC_LDS
    // Stage the raw int spikes for this chunk via the gfx1250 async global->LDS path
    // (ASYNCcnt-tracked, 512 B per b128 instruction), then convert int->f32 in LDS.
    __shared__ int s_raw[K_PER];         // 2 KB
    if (threadIdx.x < (K_PER * 4 / 16)) {        // 128 threads x 16 B = 2 KB
        gptr_v4i src = (gptr_v4i)(spikes_in + k0) + threadIdx.x;
        lptr_v4i dst = (lptr_v4i)(s_raw) + threadIdx.x;
        __builtin_amdgcn_global_load_async_to_lds_b128(src, dst, 0, 0);
#if defined(__has_builtin) && __has_builtin(__builtin_amdgcn_s_wait_asynccnt)
        __builtin_amdgcn_s_wait_asynccnt(0);
#else
        asm volatile("s_wait_asynccnt 0x0" ::: "memory");
#endif
    }
    __syncthreads();
    for (int i = threadIdx.x; i < K_PER; i += 256) s_spk[i] = (float)s_raw[i];
    __syncthreads();
#else
    for (int i = threadIdx.x; i < K_PER; i += 256) s_spk[i] = (float)spikes_in[k0 + i];
    __syncthreads();
#endif

    const int lane = threadIdx.x & 31;             // wave32
    const int wave = threadIdx.x >> 5;             // 8 waves / block
    const int n0   = blockIdx.x * COLS_PER_BLOCK + wave * 16;
    const int col  = n0 + (lane & 15);             // N index for this lane
    const int koff = (lane >> 4) << 1;             // half 0 -> K{0,1}, half 1 -> K{2,3}

    // 4 independent accumulators to pipeline the WMMA accumulate chain.
    v8f c0 = {}; v8f c1 = {}; v8f c2 = {}; v8f c3 = {};

    const float* __restrict__ wcol = weights + (size_t)k0 * NEU_OUT + col;

    for (int k = 0; k < K_PER; k += 16) {
        {
            const int kb = k + koff;
            v2f a, b;
            a.x = s_spk[kb];                // A rows replicated -> every D row = syn^T
            a.y = s_spk[kb + 1];
            const float* wp = wcol + (size_t)kb * NEU_OUT;
            b.x = wp[0];
            b.y = wp[NEU_OUT];
            c0 = __builtin_amdgcn_wmma_f32_16x16x4_f32(false, a, false, b,
                                                       (short)0, c0, false, false);
        }
        {
            const int kb = k + 4 + koff;
            v2f a, b;
            a.x = s_spk[kb];
            a.y = s_spk[kb + 1];
            const float* wp = wcol + (size_t)kb * NEU_OUT;
            b.x = wp[0];
            b.y = wp[NEU_OUT];
            c1 = __builtin_amdgcn_wmma_f32_16x16x4_f32(false, a, false, b,
                                                       (short)0, c1, false, false);
        }
        {
            const int kb = k + 8 + koff;
            v2f a, b;
            a.x = s_spk[kb];
            a.y = s_spk[kb + 1];
            const float* wp = wcol + (size_t)kb * NEU_OUT;
            b.x = wp[0];
            b.y = wp[NEU_OUT];
            c2 = __builtin_amdgcn_wmma_f32_16x16x4_f32(false, a, false, b,
                                                       (short)0, c2, false, false);
        }
        {
            const int kb = k + 12 + koff;
            v2f a, b;
            a.x = s_spk[kb];
            a.y = s_spk[kb + 1];
            const float* wp = wcol + (size_t)kb * NEU_OUT;
            b.x = wp[0];
            b.y = wp[NEU_OUT];
            c3 = __builtin_amdgcn_wmma_f32_16x16x4_f32(false, a, false, b,
                                                       (short)0, c3, false, false);
        }
    }

    // Every lane's accumulator element 0 holds syn for column n0 + lane%16
    // (all 16 M-rows identical). Lanes 0..15 publish the 16-column tile.
    const float syn = c0[0] + c1[0] + c2[0] + c3[0];
    if (lane < 16) {
        partial[blockIdx.y * NEU_OUT + col] = syn;
    }
}

__global__ __launch_bounds__(256) void snn_epilogue_kernel(
    const float* __restrict__ partial,   // [K_CHUNKS][NEU_OUT]
    const float* __restrict__ mempot,    // [NEU_OUT]
    float* __restrict__ out)             // [3*NEU_OUT]: spikes | traces | mempot_new
{
    const int col = blockIdx.x * 256 + threadIdx.x;
    float syn = 0.0f;
#pragma unroll
    for (int c = 0; c < K_CHUNKS; ++c) syn += partial[c * NEU_OUT + col];

    const float m     = mempot[col] + syn;
    const bool  fired = (m >= 1.0f);
    out[col]               = fired ? 1.0f : 0.0f;     // spikes_out
    out[NEU_OUT + col]     = 0.0f;                     // traces_out
    out[2 * NEU_OUT + col] = fired ? (m - 1.0f)        // mempot_new
                                   : (m * 0.9375f);    // *(2^4-1)/2^4
}

extern "C" void kernel_launch(void* const* d_in, const int* in_sizes, int n_in,
                              void* d_out, int out_size, void* d_ws, size_t ws_size,
                              hipStream_t stream) {
    (void)in_sizes; (void)n_in; (void)ws_size; (void)out_size;
    const int*   spikes  = (const int*)d_in[0];
    const float* weights = (const float*)d_in[1];
    const float* mempot  = (const float*)d_in[2];
    float*       out     = (float*)d_out;
    float*       partial = (float*)d_ws;   // K_CHUNKS * NEU_OUT floats = 256 KB

    // Phase 1: 64 column-tiles x 8 K-chunks = 512 blocks (4096 wave32s) -> enough
    // outstanding loads to approach 23.3 TB/s on the single-use 128 MB W stream.
    dim3 grid1(NEU_OUT / COLS_PER_BLOCK, K_CHUNKS);
    snn_partial_wmma_kernel<<<grid1, 256, 0, stream>>>(spikes, weights, partial);

    // Phase 2: deterministic split-K reduction + fire/decay epilogue.
    snn_epilogue_kernel<<<NEU_OUT / 256, 256, 0, stream>>>(partial, mempot, out);
}